// OccupancyGrid_9414568313107
// MI455X (gfx1250) — compile-verified
//
#include <hip/hip_runtime.h>
#include <math.h>

// ---- constants from the reference ----
#define N_RAYS_   32768
#define MAX_N_    512
#define N_LEVELS_ 4
#define RES_      128
#define STEP_     0.02f
#define EPS_      1.0e-4f
#define FAR_PLANE_ 1.0e10f

#define LOG2E_ 1.44269504088896340736f
#define LN2_   0.69314718055994530942f

typedef __attribute__((ext_vector_type(2))) float v2f;
typedef __attribute__((ext_vector_type(8))) float v8f;

// DPP helper: v += shifted-by-N-within-row(v), zero shifted in (bound_ctrl).
#define DPP_ADD_SHR(v, N)                                                     \
    (v) += __int_as_float(__builtin_amdgcn_update_dpp(                        \
        0, __float_as_int(v), 0x110 | (N), 0xF, 0xF, true))

// One wave32 per ray. 16 chunks of 32 samples.
//  * pos@w+b via two V_WMMA_F32_16X16X4_F32 per chunk:
//      A[m][k] = wext[k] for all rows  (lanes<16:(w0,w1), lanes>=16:(w2,b))
//      B[k][n] = posext[n][k]; D[m][n] = dot(sample n) replicated over m,
//      so each lane reads its own result from d[0] statically.
//    B-fragment halves are *recomputed* locally (2-3 FMAs) instead of
//    shuffled -> zero DS ops in the loop.
//  * Prefix scan: 4 DPP row_shr adds + readlane(15) fixup + readlane(31)
//    carry -> pure VALU, no LDS round-trips.
//  * Transcendentals: raw v_exp_f32 / v_log_f32 (TRANS, co-executes with
//    VALU); log argument is in (1,2] so no fix-up paths are needed.
__global__ __launch_bounds__(256) void occgrid_march_kernel(
    const float* __restrict__ origins,
    const float* __restrict__ viewdirs,
    const float* __restrict__ wvec,
    const float* __restrict__ bscal,
    const unsigned char* __restrict__ occ,   // bool grid, 4*128^3 bytes
    const float* __restrict__ aabbs,         // (4,6)
    float* __restrict__ out)                 // (N_RAYS, MAX_N)
{
    const int lane = threadIdx.x & 31;
    const int ray  = blockIdx.x * (blockDim.x >> 5) + (threadIdx.x >> 5);
    if (ray >= N_RAYS_) return;   // uniform per wave; never taken with exact grid

    // ---- per-ray setup (uniform across the wave) ----
    const float ox = origins[3 * ray + 0];
    const float oy = origins[3 * ray + 1];
    const float oz = origins[3 * ray + 2];
    const float dx = viewdirs[3 * ray + 0];
    const float dy = viewdirs[3 * ray + 1];
    const float dz = viewdirs[3 * ray + 2];
    const float w0 = wvec[0], w1 = wvec[1], w2 = wvec[2];
    const float bb = bscal[0];

    // level-0 box (for pos_unit) and last-level box (for ray clipping)
    const float a0minx = aabbs[0], a0miny = aabbs[1], a0minz = aabbs[2];
    const float a0maxx = aabbs[3], a0maxy = aabbs[4], a0maxz = aabbs[5];
    const float* al = aabbs + 6 * (N_LEVELS_ - 1);
    const float lminx = al[0], lminy = al[1], lminz = al[2];
    const float lmaxx = al[3], lmaxy = al[4], lmaxz = al[5];

    const float t1x = (lminx - ox) / dx, t2x = (lmaxx - ox) / dx;
    const float t1y = (lminy - oy) / dy, t2y = (lmaxy - oy) / dy;
    const float t1z = (lminz - oz) / dz, t2z = (lmaxz - oz) / dz;
    const float tmn = fmaxf(fmaxf(fminf(t1x, t2x), fminf(t1y, t2y)), fminf(t1z, t2z));
    const float tmx = fminf(fminf(fmaxf(t1x, t2x), fmaxf(t1y, t2y)), fmaxf(t1z, t2z));
    const bool  hit  = (tmx > tmn) && (tmx > 0.0f);
    const float nearp = hit ? fmaxf(tmn, 0.0f) : 0.0f;
    const float farp  = hit ? fminf(tmx, FAR_PLANE_) : 0.0f;

    const float inx = 1.0f / (a0maxx - a0minx);
    const float iny = 1.0f / (a0maxy - a0miny);
    const float inz = 1.0f / (a0maxz - a0minz);

    // A fragment: every row of A == (w0,w1,w2,b). Constant over the loop.
    const bool lo16 = (lane < 16);
    v2f afrag;
    afrag[0] = lo16 ? w0 : w2;
    afrag[1] = lo16 ? w1 : bb;
    v8f czero = {};

    float carry = 0.0f;                     // running cumsum of sigmas_dt
    float* __restrict__ orow = out + (size_t)ray * MAX_N_;

    #pragma unroll 2
    for (int chunk = 0; chunk < MAX_N_ / 32; ++chunk) {
        const int   i      = chunk * 32 + lane;
        const float fi     = (float)i;
        const float tstart = nearp + fi * STEP_;
        const float tend   = tstart + STEP_;
        const bool  in_range = (tend <= farp) && hit;
        const float tmid   = tstart + 0.5f * STEP_;

        // own sample position
        const float px = fmaf(dx, tmid, ox);
        const float py = fmaf(dy, tmid, oy);
        const float pz = fmaf(dz, tmid, oz);

        // ---- occupancy lookup ----
        const float pux = (px - a0minx) * inx;
        const float puy = (py - a0miny) * iny;
        const float puz = (pz - a0minz) * inz;
        const float mx = fmaxf(fmaxf(fabsf(pux - 0.5f), fabsf(puy - 0.5f)),
                               fabsf(puz - 0.5f));
        int e; (void)frexpf(mx, &e);        // v_frexp_exp_i32_f32
        const int  mip   = e + 1;
        const bool valid = mip < N_LEVELS_;
        const int  mipc  = mip < 0 ? 0 : (mip > N_LEVELS_ - 1 ? N_LEVELS_ - 1 : mip);
        const float scale = ldexpf(1.0f, -mipc);   // exact 2^-mipc
        const float qx = fminf(fmaxf(fmaf(pux - 0.5f, scale, 0.5f), 0.0f), 1.0f - 1e-5f);
        const float qy = fminf(fmaxf(fmaf(puy - 0.5f, scale, 0.5f), 0.0f), 1.0f - 1e-5f);
        const float qz = fminf(fmaxf(fmaf(puz - 0.5f, scale, 0.5f), 0.0f), 1.0f - 1e-5f);
        const int ix = (int)(qx * (float)RES_);
        const int iy = (int)(qy * (float)RES_);
        const int iz = (int)(qz * (float)RES_);
        const bool occ_hit =
            occ[(size_t)(((mipc * RES_ + ix) * RES_ + iy) * RES_ + iz)] != 0;
        const bool mask = in_range && valid && occ_hit;

        // ---- B-fragments: recompute the opposite half's position locally ----
        // lane<16 needs pos of sample i+16 (x,y); lane>=16 needs pos of
        // sample i-16 (z). Same arithmetic expressions -> identical values.
        const float fio  = lo16 ? (fi + 16.0f) : (fi - 16.0f);
        const float tso  = nearp + fio * STEP_;
        const float tmo  = tso + 0.5f * STEP_;
        const float bx_o = fmaf(dx, tmo, ox);
        const float by_o = fmaf(dy, tmo, oy);
        const float bz_o = fmaf(dz, tmo, oz);

        v2f b1, b2;
        b1[0] = lo16 ? px   : bz_o;  b1[1] = lo16 ? py   : 1.0f;
        b2[0] = lo16 ? bx_o : pz;    b2[1] = lo16 ? by_o : 1.0f;
        v8f d1 = __builtin_amdgcn_wmma_f32_16x16x4_f32(
            false, afrag, false, b1, (short)0, czero, false, false);
        v8f d2 = __builtin_amdgcn_wmma_f32_16x16x4_f32(
            false, afrag, false, b2, (short)0, czero, false, false);
        const float xv = lo16 ? d1[0] : d2[0];

        // stable softplus via raw exp2/log2 (log arg in (1,2], no fix-ups)
        const float e1 = __builtin_amdgcn_exp2f(fabsf(xv) * -LOG2E_);
        const float sp = fmaxf(xv, 0.0f) + __builtin_amdgcn_logf(1.0f + e1) * LN2_;
        const float sdt = mask ? sp * STEP_ : 0.0f;

        // ---- wave inclusive scan: DPP within rows, readlane across ----
        float scn = sdt;
        DPP_ADD_SHR(scn, 1);
        DPP_ADD_SHR(scn, 2);
        DPP_ADD_SHR(scn, 4);
        DPP_ADD_SHR(scn, 8);
        const float row0sum =
            __int_as_float(__builtin_amdgcn_readlane(__float_as_int(scn), 15));
        const float incl = scn + (lo16 ? 0.0f : row0sum);
        const float total =
            __int_as_float(__builtin_amdgcn_readlane(__float_as_int(incl), 31));

        const float excl  = carry + incl - sdt;   // exclusive prefix + carry
        const float trans = __builtin_amdgcn_exp2f(excl * -LOG2E_);
        const float alpha = 1.0f - __builtin_amdgcn_exp2f(sdt * -LOG2E_);
        // ALPHA_THRE == 0 and alpha >= 0 always; only T >= EPS and mask matter.
        const bool visible = (trans >= EPS_) && mask;
        orow[i] = visible ? alpha * trans : 0.0f;

        carry += total;                           // uniform chunk total
    }
}

extern "C" void kernel_launch(void* const* d_in, const int* in_sizes, int n_in,
                              void* d_out, int out_size, void* d_ws, size_t ws_size,
                              hipStream_t stream) {
    const float*         origins  = (const float*)d_in[0];
    const float*         viewdirs = (const float*)d_in[1];
    const float*         wvec     = (const float*)d_in[2];
    const float*         bscal    = (const float*)d_in[3];
    const unsigned char* occ      = (const unsigned char*)d_in[4];  // bool grid
    const float*         aabbs    = (const float*)d_in[5];
    float*               out      = (float*)d_out;

    const int threads = 256;                        // 8 wave32 = 8 rays / block
    const int blocks  = N_RAYS_ / (threads / 32);   // 4096
    occgrid_march_kernel<<<blocks, threads, 0, stream>>>(
        origins, viewdirs, wvec, bscal, occ, aabbs, out);
}